// HGTLayer_23708219474683
// MI455X (gfx1250) — compile-verified
//
#include <hip/hip_runtime.h>

typedef float v2f __attribute__((ext_vector_type(2)));
typedef float v8f __attribute__((ext_vector_type(8)));

#define DIMC 128
#define HEADSC 8
#define DKC 16

// ---------------------------------------------------------------------------
// Fold per-head relation matrices into projection weights:
// Weff[i][h*16+j] = sum_t Wsrc[i][h*16+t] * rel[h][t][j]   (also for bias row)
// ---------------------------------------------------------------------------
__global__ __launch_bounds__(256) void prep_weights(
    const float* __restrict__ Wsrc, const float* __restrict__ bsrc,
    const float* __restrict__ rel, float* __restrict__ Weff,
    float* __restrict__ beff) {
  int idx = blockIdx.x * blockDim.x + threadIdx.x;
  if (idx >= (DIMC + 1) * DIMC) return;
  int i = idx >> 7;                 // 0..128 (128 == bias row)
  int c = idx & (DIMC - 1);
  int hh = c >> 4, j = c & 15;
  const float* r = rel + hh * (DKC * DKC) + j;   // rel[h][t][j], t-stride 16
  float s = 0.f;
  if (i < DIMC) {
    const float* w = Wsrc + i * DIMC + hh * DKC;
#pragma unroll
    for (int t = 0; t < DKC; ++t) s += w[t] * r[t * DKC];
    Weff[i * DIMC + c] = s;
  } else {
    const float* w = bsrc + hh * DKC;
#pragma unroll
    for (int t = 0; t < DKC; ++t) s += w[t] * r[t * DKC];
    beff[c] = s;
  }
}

// ---------------------------------------------------------------------------
// Zero accumulators; smax gets 0x00000000 == order-encoded -inf
// ---------------------------------------------------------------------------
__global__ __launch_bounds__(256) void init_buffers(
    float* __restrict__ t, float* __restrict__ ssum,
    unsigned* __restrict__ smax, int N) {
  int i = blockIdx.x * blockDim.x + threadIdx.x;
  if (i < N * DIMC) t[i] = 0.f;
  if (i < N * HEADSC) { ssum[i] = 0.f; smax[i] = 0u; }
}

// ---------------------------------------------------------------------------
// out[N,128] = X[N,128] @ W[128,128] + bias   (blend: sigmoid-skip with h)
// One wave = one 16-col strip; 8 waves/block cover all 128 cols of a 16-row
// tile. K loop = 32 x v_wmma_f32_16x16x4_f32. B fragments preloaded once per
// wave and reused over M tiles (W stays in VGPRs, X rows stream from L2).
// ---------------------------------------------------------------------------
__global__ __launch_bounds__(256) void wmma_gemm128(
    const float* __restrict__ X, const float* __restrict__ W,
    const float* __restrict__ bias, const float* __restrict__ hres,
    const float* __restrict__ skip, float* __restrict__ out,
    int N, int blend) {
  const int lane  = threadIdx.x & 31;
  const int wave  = threadIdx.x >> 5;   // 0..7
  const int col0  = wave * 16;
  const int halfK = lane >> 4;          // 0: K 0..1, 1: K 2..3 of each step
  const int ln    = lane & 15;

  // B fragment per K-step: lane holds W[k + halfK*2 + {0,1}][col0+ln]
  v2f Bfrag[32];
#pragma unroll
  for (int kk = 0; kk < 32; ++kk) {
    const float* wp = W + (kk * 4 + halfK * 2) * DIMC + col0 + ln;
    Bfrag[kk][0] = wp[0];
    Bfrag[kk][1] = wp[DIMC];
  }
  float bcol = bias[col0 + ln];
  float alpha = 1.0f, beta = 0.0f;
  if (blend) {
    float s = skip[0];
    alpha = 1.0f / (1.0f + __expf(-s));
    beta  = 1.0f - alpha;
  }

  for (int m0 = blockIdx.x * 16; m0 < N; m0 += gridDim.x * 16) {
    v8f C = {0.f, 0.f, 0.f, 0.f, 0.f, 0.f, 0.f, 0.f};
    const float* xrow = X + (size_t)(m0 + ln) * DIMC + halfK * 2;
#pragma unroll
    for (int kk = 0; kk < 32; ++kk) {
      v2f A;
      A[0] = xrow[kk * 4 + 0];
      A[1] = xrow[kk * 4 + 1];
      C = __builtin_amdgcn_wmma_f32_16x16x4_f32(
          false, A, false, Bfrag[kk], (short)0, C, false, false);
    }
    // C VGPR r: lanes 0-15 -> row m0+r, lanes 16-31 -> row m0+8+r
    int rbase = m0 + halfK * 8;
#pragma unroll
    for (int r = 0; r < 8; ++r) {
      size_t idx = (size_t)(rbase + r) * DIMC + col0 + ln;
      float val = C[r] + bcol;
      if (blend) val = alpha * val + beta * hres[idx];
      out[idx] = val;
    }
  }
}

// ---------------------------------------------------------------------------
// Per (edge, head): score = dot16(q[dst], k'[src]) * rel_pri / 4
// segment max via order-preserving uint atomicMax
// ---------------------------------------------------------------------------
__device__ __forceinline__ unsigned f32_order_encode(float f) {
  unsigned u = __float_as_uint(f);
  return (u & 0x80000000u) ? ~u : (u | 0x80000000u);
}
__device__ __forceinline__ float f32_order_decode(unsigned u) {
  return (u & 0x80000000u) ? __uint_as_float(u ^ 0x80000000u)
                           : __uint_as_float(~u);
}

__global__ __launch_bounds__(256) void edge_score(
    const float* __restrict__ q, const float* __restrict__ kmat,
    const int* __restrict__ src, const int* __restrict__ dst,
    const float* __restrict__ rel_pri, float* __restrict__ score,
    unsigned* __restrict__ smax, int E) {
  int i = blockIdx.x * blockDim.x + threadIdx.x;
  if (i >= E * HEADSC) return;
  int e = i >> 3, hh = i & 7;
  int s = src[e], d = dst[e];
  const float4* qp = (const float4*)(q    + (size_t)d * DIMC + hh * DKC);
  const float4* kp = (const float4*)(kmat + (size_t)s * DIMC + hh * DKC);
  float acc = 0.f;
#pragma unroll
  for (int c = 0; c < 4; ++c) {
    float4 a = qp[c], b = kp[c];
    acc += a.x * b.x + a.y * b.y + a.z * b.z + a.w * b.w;
  }
  float sc = acc * rel_pri[hh] * 0.25f;   // 1/sqrt(16)
  score[i] = sc;
  atomicMax(&smax[(size_t)d * HEADSC + hh], f32_order_encode(sc));
}

// ---------------------------------------------------------------------------
// Per (edge, head): e = exp(score - smax[dst]); ssum[dst] += e;
// t[dst, head, :] += e * v'[src, head, :]   (unnormalized; divide later)
// ---------------------------------------------------------------------------
__global__ __launch_bounds__(256) void edge_accum(
    const float* __restrict__ score, const unsigned* __restrict__ smax,
    const float* __restrict__ v, const int* __restrict__ src,
    const int* __restrict__ dst, float* __restrict__ ssum,
    float* __restrict__ t, int E) {
  int i = blockIdx.x * blockDim.x + threadIdx.x;
  if (i >= E * HEADSC) return;
  int e = i >> 3, hh = i & 7;
  int s = src[e], d = dst[e];
  float m  = f32_order_decode(smax[(size_t)d * HEADSC + hh]);
  float ee = __expf(score[i] - m);
  atomicAdd(&ssum[(size_t)d * HEADSC + hh], ee);
  const float* vp = v + (size_t)s * DIMC + hh * DKC;
  float*       tp = t + (size_t)d * DIMC + hh * DKC;
#pragma unroll
  for (int c = 0; c < DKC; ++c) atomicAdd(&tp[c], ee * vp[c]);
}

// ---------------------------------------------------------------------------
// t[n, h*16+d] /= ssum[n, h]   (guard edge-less nodes: 0/0 -> 0)
// ---------------------------------------------------------------------------
__global__ __launch_bounds__(256) void normalize_t(
    float* __restrict__ t, const float* __restrict__ ssum, int n) {
  int i = blockIdx.x * blockDim.x + threadIdx.x;
  if (i >= n) return;
  float s = ssum[(size_t)(i >> 7) * HEADSC + ((i & 127) >> 4)];
  t[i] = (s > 0.f) ? t[i] / s : 0.f;
}

// ---------------------------------------------------------------------------
extern "C" void kernel_launch(void* const* d_in, const int* in_sizes, int n_in,
                              void* d_out, int out_size, void* d_ws,
                              size_t ws_size, hipStream_t stream) {
  const float* h       = (const float*)d_in[0];
  const int*   src     = (const int*)  d_in[1];
  const int*   dst     = (const int*)  d_in[2];
  const float* Wk      = (const float*)d_in[3];
  const float* bk      = (const float*)d_in[4];
  const float* Wq      = (const float*)d_in[5];
  const float* bq      = (const float*)d_in[6];
  const float* Wv      = (const float*)d_in[7];
  const float* bv      = (const float*)d_in[8];
  const float* Wa      = (const float*)d_in[9];
  const float* ba      = (const float*)d_in[10];
  const float* rel_att = (const float*)d_in[11];
  const float* rel_msg = (const float*)d_in[12];
  const float* rel_pri = (const float*)d_in[13];
  const float* skip    = (const float*)d_in[14];
  float* out = (float*)d_out;

  const int N = in_sizes[0] / DIMC;
  const int E = in_sizes[1];

  // workspace carve-up (floats)
  float* ws   = (float*)d_ws;
  size_t node = (size_t)N * DIMC;
  float* q     = ws;
  float* kp    = q  + node;
  float* vp    = kp + node;
  float* t     = vp + node;
  float* score = t  + node;
  float* ssum  = score + (size_t)E * HEADSC;
  unsigned* smax = (unsigned*)(ssum + (size_t)N * HEADSC);
  float* Wke = (float*)(smax + (size_t)N * HEADSC);
  float* bke = Wke + DIMC * DIMC;
  float* Wve = bke + DIMC;
  float* bve = Wve + DIMC * DIMC;

  const int prepN = (DIMC + 1) * DIMC;
  prep_weights<<<(prepN + 255) / 256, 256, 0, stream>>>(Wk, bk, rel_att, Wke, bke);
  prep_weights<<<(prepN + 255) / 256, 256, 0, stream>>>(Wv, bv, rel_msg, Wve, bve);
  init_buffers<<<((int)node + 255) / 256, 256, 0, stream>>>(t, ssum, smax, N);

  // projections (k', v' have relation transform folded in)
  wmma_gemm128<<<625, 256, 0, stream>>>(h, Wq,  bq,  nullptr, nullptr, q,  N, 0);
  wmma_gemm128<<<625, 256, 0, stream>>>(h, Wke, bke, nullptr, nullptr, kp, N, 0);
  wmma_gemm128<<<625, 256, 0, stream>>>(h, Wve, bve, nullptr, nullptr, vp, N, 0);

  const int EH = E * HEADSC;
  edge_score<<<(EH + 255) / 256, 256, 0, stream>>>(q, kp, src, dst, rel_pri,
                                                   score, smax, E);
  edge_accum<<<(EH + 255) / 256, 256, 0, stream>>>(score, smax, vp, src, dst,
                                                   ssum, t, E);
  normalize_t<<<((int)node + 255) / 256, 256, 0, stream>>>(t, ssum, (int)node);

  // out = sigmoid(skip) * (t @ Wa + ba) + (1 - sigmoid(skip)) * h
  wmma_gemm128<<<625, 256, 0, stream>>>(t, Wa, ba, h, skip, out, N, 1);
}